// HistogramMaskLoss_32444182954404
// MI455X (gfx1250) — compile-verified
//
#include <hip/hip_runtime.h>
#include <math.h>

#define NUM_BINS 100
#define EPS_F 1e-6f

typedef __attribute__((ext_vector_type(2))) float v2f;
typedef __attribute__((ext_vector_type(8))) float v8f;

// ---------------------------------------------------------------------------
// Kernel 0: zero the accumulation workspace (2*100 bins + 2 counts, u32)
// ---------------------------------------------------------------------------
__global__ void HistLoss_zero_ws(unsigned int* __restrict__ ws) {
    int i = threadIdx.x;
    if (i < 2 * NUM_BINS + 2) ws[i] = 0u;
}

// ---------------------------------------------------------------------------
// Kernel 1: streaming distance + histogram.
//
// Each wave owns a group of 16 consecutive pixels. Per K-chunk of 4 channels:
//   lanes 0-15  load diffs for channels (c0, c1) of pixel (lane&15)
//   lanes 16-31 load diffs for channels (c2, c3) of the same pixel
// That *is* the A-matrix layout for V_WMMA_F32_16X16X4_F32 (16x4, lanes 0-15
// hold K=0/1, lanes 16-31 hold K=2/3). The B-matrix (= A^T, rows K0..K3
// striped across lanes within a VGPR) needs one value swapped across the
// half-waves, done with ds_bpermute (lane ^ 16). Eight chained WMMAs give
// C[i][j] = sum_c diff[i][c]*diff[j][c]; the diagonal is the per-pixel
// squared L2 distance over all 32 channels.
// ---------------------------------------------------------------------------
__global__ __launch_bounds__(256) void HistLoss_hist_kernel(
    const float* __restrict__ t0, const float* __restrict__ t1,
    const int* __restrict__ gt, unsigned int* __restrict__ ws,
    int HW, int C, int nGroups)
{
    __shared__ unsigned int lh[2 * NUM_BINS + 2];
    for (int i = threadIdx.x; i < 2 * NUM_BINS + 2; i += blockDim.x) lh[i] = 0u;
    __syncthreads();

    const int lane          = threadIdx.x & 31;
    const int waveInBlock   = threadIdx.x >> 5;
    const int wavesPerBlock = blockDim.x >> 5;
    const int waveGlobal    = blockIdx.x * wavesPerBlock + waveInBlock;
    const int waveCount     = gridDim.x * wavesPerBlock;
    const bool lower        = (lane < 16);
    const int  pixLane      = lane & 15;
    const int  kChunks      = C >> 2;                 // C == 32 -> 8 chunks
    const int  permIdx      = (lane ^ 16) << 2;       // ds_bpermute byte index

    for (int g = waveGlobal; g < nGroups; g += waveCount) {
        const int base = g * 16;
        const int p    = base + pixLane;

        v8f acc = {0.f, 0.f, 0.f, 0.f, 0.f, 0.f, 0.f, 0.f};

        for (int k = 0; k < kChunks; ++k) {
            // lower half loads channels 4k,4k+1 ; upper half 4k+2,4k+3
            const int c0 = 4 * k + (lower ? 0 : 2);
            const size_t off = (size_t)c0 * (size_t)HW + (size_t)p;
            const float x0a = t0[off];
            const float x1a = t1[off];
            const float x0b = t0[off + (size_t)HW];
            const float x1b = t1[off + (size_t)HW];
            const float dA = (x0a - x1a) + EPS_F;     // diff at channel c0
            const float dB = (x0b - x1b) + EPS_F;     // diff at channel c0+1

            v2f A; A.x = dA; A.y = dB;                // A: (K0,K1)|(K2,K3)

            // B = A^T: lower lane n needs (diff(c0,n), diff(c2,n));
            // upper lane n+16 needs (diff(c1,n), diff(c3,n)).
            const float send = lower ? dB : dA;
            const float recv = __int_as_float(
                __builtin_amdgcn_ds_bpermute(permIdx, __float_as_int(send)));
            v2f B;
            B.x = lower ? dA : recv;
            B.y = lower ? recv : dB;

            acc = __builtin_amdgcn_wmma_f32_16x16x4_f32(
                /*neg_a=*/false, A, /*neg_b=*/false, B,
                /*c_mod=*/(short)0, acc, /*reuse_a=*/false, /*reuse_b=*/false);
        }

        // Diagonal of C: pixel m<8 -> lane m, VGPR m; m>=8 -> lane m+16, VGPR m-8
        const bool active = (lane < 8) || (lane >= 24);
        const int  idx    = lane & 7;
        float d2 = acc[0];
        d2 = (idx == 1) ? acc[1] : d2;
        d2 = (idx == 2) ? acc[2] : d2;
        d2 = (idx == 3) ? acc[3] : d2;
        d2 = (idx == 4) ? acc[4] : d2;
        d2 = (idx == 5) ? acc[5] : d2;
        d2 = (idx == 6) ? acc[6] : d2;
        d2 = (idx == 7) ? acc[7] : d2;

        if (active) {
            const int pixel = base + ((lane < 8) ? lane : (lane - 16));
            const float d   = sqrtf(d2);
            const bool  pos = (gt[pixel] == 0);
            atomicAdd(&lh[2 * NUM_BINS + (pos ? 0 : 1)], 1u);
            if (d >= 0.0f && d <= 1.0f) {             // histc range filter
                int b = (int)floorf(d * (float)NUM_BINS);
                b = (b < 0) ? 0 : ((b > NUM_BINS - 1) ? NUM_BINS - 1 : b);
                atomicAdd(&lh[(pos ? 0 : NUM_BINS) + b], 1u);
            }
        }
    }

    __syncthreads();
    for (int i = threadIdx.x; i < 2 * NUM_BINS + 2; i += blockDim.x) {
        const unsigned int v = lh[i];
        if (v) atomicAdd(&ws[i], v);
    }
}

// ---------------------------------------------------------------------------
// Kernel 2: normalize histograms, KL(reduction='mean'), loss = 1 + kl
// ---------------------------------------------------------------------------
__global__ void HistLoss_kl_kernel(const unsigned int* __restrict__ ws,
                                   float* __restrict__ out) {
    __shared__ float partial[128];
    const int t = threadIdx.x;
    const float posN = (float)ws[2 * NUM_BINS + 0];
    const float negN = (float)ws[2 * NUM_BINS + 1];

    float v = 0.0f;
    if (t < NUM_BINS) {
        const float hp = (posN > 0.0f) ? ((float)ws[t] / posN) : 0.0f;
        const float hn = (negN > 0.0f) ? ((float)ws[NUM_BINS + t] / negN) : 0.0f;
        if (hn > 0.0f) v = hn * (logf(hn) - hp);
    }
    partial[t] = v;
    __syncthreads();
    for (int s = 64; s > 0; s >>= 1) {
        if (t < s) partial[t] += partial[t + s];
        __syncthreads();
    }
    if (t == 0) out[0] = 1.0f + partial[0] / (float)NUM_BINS;
}

// ---------------------------------------------------------------------------
extern "C" void kernel_launch(void* const* d_in, const int* in_sizes, int n_in,
                              void* d_out, int out_size, void* d_ws, size_t ws_size,
                              hipStream_t stream) {
    const float* t0 = (const float*)d_in[0];
    const float* t1 = (const float*)d_in[1];
    const int*   gt = (const int*)d_in[2];

    const int HW = in_sizes[2];              // n*h*w  (n==1)
    const int C  = in_sizes[0] / HW;         // 32
    const int nGroups = HW / 16;             // 16 pixels per wave-group

    unsigned int* ws = (unsigned int*)d_ws;

    HistLoss_zero_ws<<<1, 256, 0, stream>>>(ws);

    // 1024 blocks x 8 waves = 8192 waves; 65536 groups -> 8 groups/wave.
    HistLoss_hist_kernel<<<1024, 256, 0, stream>>>(t0, t1, gt, ws, HW, C, nGroups);

    HistLoss_kl_kernel<<<1, 128, 0, stream>>>(ws, (float*)d_out);
}